// CrossInvariantPointAttention_57664230916411
// MI455X (gfx1250) — compile-verified
//
#include <hip/hip_runtime.h>
#include <hip/hip_bf16.h>
#include <math.h>

// ---------------- problem constants ----------------
#define B_   4
#define LD_  512
#define LS_  512
#define CS_  384
#define CH_  16
#define H_   12
#define PQ_  4
#define PV_  8
#define EPS_ 1e-8f
#define INF_ 100000.0f
// W_C = sqrt(2/(9*PQ)) , W_L = sqrt(0.5)
#define W_C_ 0.2357022603955158f
#define W_L_ 0.7071067811865476f

typedef __attribute__((ext_vector_type(2))) float v2f;
typedef __attribute__((ext_vector_type(8))) float v8f;

// =====================================================================
// Register-blocked batched fp32 WMMA GEMM: C[b] = A[b] * B[b] (+ bias)
// One wave computes an (MT*16) x (NT*16) tile with MT*NT independent
// V_WMMA_F32_16X16X4_F32 accumulator chains (ILP across the matrix
// pipe's D->C RAW latency) and MT+NT fragment loads per k-step
// (2x operand reuse at <2,2> vs the naive 1-tile version).
//
// fragment layout (CDNA5 ISA 7.12.2, 32-bit):
//   A 16x4 : lanes 0-15 -> rows M=0..15 with K=0(v0),K=1(v1);
//            lanes 16-31 -> same rows, K=2(v0),K=3(v1)
//   B 4x16 : lanes 0-15 -> cols N=0..15 with K=0(v0),K=1(v1);
//            lanes 16-31 -> same cols, K=2(v0),K=3(v1)
//   C/D    : VGPR r -> row (r + 8*laneHalf), col = lane&15
// Requires M % (MT*16) == 0, N % (NT*16) == 0, K % 4 == 0 (all grids
// below are chosen to divide exactly -> EXEC stays all-ones).
// =====================================================================
template <int MT, int NT>
__global__ __launch_bounds__(32)
void wmma_gemm_f32(const float* __restrict__ A, int lda, long long strideA,
                   const float* __restrict__ Bm, int ldb, long long strideB,
                   float* __restrict__ C, int ldc, long long strideC,
                   int K, const float* __restrict__ bias)
{
    const int lane = threadIdx.x;          // 0..31
    const int half = lane >> 4;            // 0 or 1
    const int l    = lane & 15;
    const int tn   = blockIdx.x * (16 * NT);
    const int tm   = blockIdx.y * (16 * MT);
    const long long batch = blockIdx.z;

    const float* a_row[MT];
    const float* b_col[NT];
#pragma unroll
    for (int mi = 0; mi < MT; ++mi)
        a_row[mi] = A + batch * strideA + (long long)(tm + mi * 16 + l) * lda + half * 2;
#pragma unroll
    for (int ni = 0; ni < NT; ++ni)
        b_col[ni] = Bm + batch * strideB + (long long)(half * 2) * ldb + (tn + ni * 16 + l);

    v8f acc[MT][NT];
#pragma unroll
    for (int mi = 0; mi < MT; ++mi)
#pragma unroll
        for (int ni = 0; ni < NT; ++ni)
            acc[mi][ni] = (v8f){};

    for (int k = 0; k < K; k += 4) {
        v2f av[MT], bv[NT];
#pragma unroll
        for (int mi = 0; mi < MT; ++mi) {
            av[mi].x = a_row[mi][k];
            av[mi].y = a_row[mi][k + 1];
        }
#pragma unroll
        for (int ni = 0; ni < NT; ++ni) {
            bv[ni].x = b_col[ni][(long long)k * ldb];
            bv[ni].y = b_col[ni][(long long)(k + 1) * ldb];
        }
#pragma unroll
        for (int mi = 0; mi < MT; ++mi)
#pragma unroll
            for (int ni = 0; ni < NT; ++ni)
                acc[mi][ni] = __builtin_amdgcn_wmma_f32_16x16x4_f32(
                                  /*neg_a=*/false, av[mi], /*neg_b=*/false, bv[ni],
                                  /*c_mod=*/(short)0, acc[mi][ni],
                                  /*reuse_a=*/false, /*reuse_b=*/false);
    }

#pragma unroll
    for (int mi = 0; mi < MT; ++mi) {
        float* c_base = C + batch * strideC + (long long)(tm + mi * 16 + half * 8) * ldc;
#pragma unroll
        for (int ni = 0; ni < NT; ++ni) {
            const int col = tn + ni * 16 + l;
            const float bb = bias ? bias[col] : 0.0f;
#pragma unroll
            for (int r = 0; r < 8; ++r)
                c_base[(long long)r * ldc + col] = acc[mi][ni][r] + bb;
        }
    }
}

__device__ __forceinline__ float softplusf(float x) { return log1pf(expf(x)); }

// =====================================================================
// Build Qext (B,H,LD,32): [ q*sA | q_pts_glob*sP | -0.5*sP*|q_pts|^2 | 1 | 0 0 ]
// =====================================================================
__global__ void build_q_kernel(const float* __restrict__ Qraw,   // (B*LD,192)
                               const float* __restrict__ QPraw,  // (B*LD,144) d-major
                               const float* __restrict__ Rdst,   // (B*LD,3,3)
                               const float* __restrict__ tdst,   // (B*LD,3)
                               const float* __restrict__ head_w,
                               float* __restrict__ Qext)
{
    int idx = blockIdx.x * blockDim.x + threadIdx.x;
    if (idx >= B_ * LD_ * H_) return;
    int h = idx % H_;
    int i = (idx / H_) % LD_;
    int b = idx / (H_ * LD_);
    long long row = (long long)b * LD_ + i;

    const float* R = Rdst + row * 9;
    const float* t = tdst + row * 3;
    float hw = softplusf(head_w[h]);
    float sA = W_L_ / sqrtf((float)CH_);
    float sP = W_L_ * W_C_ * hw;

    float* out = Qext + (((long long)(b * H_ + h) * LD_) + i) * 32;
    const float* q = Qraw + row * (H_ * CH_) + h * CH_;
#pragma unroll
    for (int c = 0; c < CH_; ++c) out[c] = q[c] * sA;

    float q2 = 0.0f;
#pragma unroll
    for (int p = 0; p < PQ_; ++p) {
        float lx = QPraw[row * 144 + 0 * 48 + h * PQ_ + p];
        float ly = QPraw[row * 144 + 1 * 48 + h * PQ_ + p];
        float lz = QPraw[row * 144 + 2 * 48 + h * PQ_ + p];
        float gx = R[0] * lx + R[1] * ly + R[2] * lz + t[0];
        float gy = R[3] * lx + R[4] * ly + R[5] * lz + t[1];
        float gz = R[6] * lx + R[7] * ly + R[8] * lz + t[2];
        out[16 + p * 3 + 0] = gx * sP;
        out[16 + p * 3 + 1] = gy * sP;
        out[16 + p * 3 + 2] = gz * sP;
        q2 += gx * gx + gy * gy + gz * gz;
    }
    out[28] = -0.5f * sP * q2;
    out[29] = 1.0f;
    out[30] = 0.0f;
    out[31] = 0.0f;
}

// =====================================================================
// Build KextT (B,H,32,LS) and VC (B,H,LS,48) = [ v | v_pts_glob | pad0 ]
// KextT rows: [ k | k_pts_glob | 1 | -0.5*sP*|k_pts|^2 | 0 0 ]^T
// =====================================================================
__global__ void build_k_kernel(const float* __restrict__ KVraw,   // (B*LS,384)
                               const float* __restrict__ KVPraw,  // (B*LS,432) d-major
                               const float* __restrict__ Rsrc,
                               const float* __restrict__ tsrc,
                               const float* __restrict__ head_w,
                               float* __restrict__ KextT,
                               float* __restrict__ VC)
{
    int idx = blockIdx.x * blockDim.x + threadIdx.x;
    if (idx >= B_ * LS_ * H_) return;
    int h = idx % H_;
    int j = (idx / H_) % LS_;
    int b = idx / (H_ * LS_);
    long long row = (long long)b * LS_ + j;

    const float* R = Rsrc + row * 9;
    const float* t = tsrc + row * 3;
    float hw = softplusf(head_w[h]);
    float sP = W_L_ * W_C_ * hw;

    const float* kv = KVraw + row * (H_ * 2 * CH_) + h * 2 * CH_;
    float* kt = KextT + (long long)(b * H_ + h) * 32 * LS_;   // [kk][j]
    float* vc = VC + (((long long)(b * H_ + h) * LS_) + j) * 48;

#pragma unroll
    for (int c = 0; c < CH_; ++c) {
        kt[c * LS_ + j] = kv[c];
        vc[c] = kv[CH_ + c];
    }

    float k2 = 0.0f;
#pragma unroll
    for (int p = 0; p < PQ_ + PV_; ++p) {
        float lx = KVPraw[row * 432 + 0 * 144 + h * 12 + p];
        float ly = KVPraw[row * 432 + 1 * 144 + h * 12 + p];
        float lz = KVPraw[row * 432 + 2 * 144 + h * 12 + p];
        float gx = R[0] * lx + R[1] * ly + R[2] * lz + t[0];
        float gy = R[3] * lx + R[4] * ly + R[5] * lz + t[1];
        float gz = R[6] * lx + R[7] * ly + R[8] * lz + t[2];
        if (p < PQ_) {
            kt[(16 + p * 3 + 0) * LS_ + j] = gx;
            kt[(16 + p * 3 + 1) * LS_ + j] = gy;
            kt[(16 + p * 3 + 2) * LS_ + j] = gz;
            k2 += gx * gx + gy * gy + gz * gz;
        } else {
            int q = p - PQ_;
            vc[16 + q * 3 + 0] = gx;
            vc[16 + q * 3 + 1] = gy;
            vc[16 + q * 3 + 2] = gz;
        }
    }
    kt[28 * LS_ + j] = 1.0f;
    kt[29 * LS_ + j] = -0.5f * sP * k2;
    kt[30 * LS_ + j] = 0.0f;
    kt[31 * LS_ + j] = 0.0f;
#pragma unroll
    for (int c = 40; c < 48; ++c) vc[c] = 0.0f;   // N-padding
}

// =====================================================================
// In-place masked softmax over last dim; one block per (b,h,i) row.
// =====================================================================
__global__ __launch_bounds__(256)
void softmax_kernel(float* __restrict__ a,
                    const unsigned char* __restrict__ dst_mask,
                    const unsigned char* __restrict__ src_mask)
{
    const int rowid = blockIdx.x;            // (b*H + h)*LD + i
    const int i  = rowid % LD_;
    const int bh = rowid / LD_;
    const int b  = bh / H_;
    float* row = a + (long long)rowid * LS_;
    const bool dm = dst_mask[b * LD_ + i] != 0;

    __shared__ float red[256];
    const int tid = threadIdx.x;

    float vals[2];
    float m = -3.4e38f;
#pragma unroll
    for (int u = 0; u < 2; ++u) {
        int j = tid + u * 256;
        float v = row[j];
        bool sm = src_mask[b * LS_ + j] != 0;
        if (!(dm && sm)) v -= INF_;
        vals[u] = v;
        m = fmaxf(m, v);
    }
    red[tid] = m; __syncthreads();
    for (int s = 128; s > 0; s >>= 1) {
        if (tid < s) red[tid] = fmaxf(red[tid], red[tid + s]);
        __syncthreads();
    }
    m = red[0]; __syncthreads();

    float sum = 0.0f;
#pragma unroll
    for (int u = 0; u < 2; ++u) { vals[u] = expf(vals[u] - m); sum += vals[u]; }
    red[tid] = sum; __syncthreads();
    for (int s = 128; s > 0; s >>= 1) {
        if (tid < s) red[tid] += red[tid + s];
        __syncthreads();
    }
    float inv = 1.0f / red[0];
#pragma unroll
    for (int u = 0; u < 2; ++u) row[tid + u * 256] = vals[u] * inv;
}

// =====================================================================
// Build cat (B*LD,576) from OC (B,H,LD,48): inverse-rotate o_pt, norms.
// cat = [ o(192) | x(96) | y(96) | z(96) | norm(96) ]
// =====================================================================
__global__ void build_cat_kernel(const float* __restrict__ OC,
                                 const float* __restrict__ Rdst,
                                 const float* __restrict__ tdst,
                                 float* __restrict__ Cat)
{
    int idx = blockIdx.x * blockDim.x + threadIdx.x;
    if (idx >= B_ * LD_ * H_) return;
    int h = idx % H_;
    int i = (idx / H_) % LD_;
    int b = idx / (H_ * LD_);
    long long row = (long long)b * LD_ + i;

    const float* oc = OC + (((long long)(b * H_ + h) * LD_) + i) * 48;
    const float* R  = Rdst + row * 9;
    const float* t  = tdst + row * 3;
    float* cat = Cat + row * 576;

#pragma unroll
    for (int c = 0; c < CH_; ++c) cat[h * CH_ + c] = oc[c];

#pragma unroll
    for (int p = 0; p < PV_; ++p) {
        float gx = oc[16 + p * 3 + 0] - t[0];
        float gy = oc[16 + p * 3 + 1] - t[1];
        float gz = oc[16 + p * 3 + 2] - t[2];
        // R^T * g  (einsum 'blji,...j->...i')
        float lx = R[0] * gx + R[3] * gy + R[6] * gz;
        float ly = R[1] * gx + R[4] * gy + R[7] * gz;
        float lz = R[2] * gx + R[5] * gy + R[8] * gz;
        float nrm = sqrtf(lx * lx + ly * ly + lz * lz + EPS_);
        cat[192 + h * PV_ + p] = lx;
        cat[288 + h * PV_ + p] = ly;
        cat[384 + h * PV_ + p] = lz;
        cat[480 + h * PV_ + p] = nrm;
    }
}

// =====================================================================
// Host-side orchestration
// =====================================================================
extern "C" void kernel_launch(void* const* d_in, const int* in_sizes, int n_in,
                              void* d_out, int out_size, void* d_ws, size_t ws_size,
                              hipStream_t stream)
{
    (void)in_sizes; (void)n_in; (void)out_size; (void)ws_size;

    const float* s_dst  = (const float*)d_in[0];
    const float* s_src  = (const float*)d_in[1];
    const float* R_dst  = (const float*)d_in[2];
    const float* t_dst  = (const float*)d_in[3];
    const float* R_src  = (const float*)d_in[4];
    const float* t_src  = (const float*)d_in[5];
    const unsigned char* dst_mask = (const unsigned char*)d_in[6];
    const unsigned char* src_mask = (const unsigned char*)d_in[7];
    const float* Wq     = (const float*)d_in[8];
    const float* Wkv    = (const float*)d_in[9];
    const float* Wqp    = (const float*)d_in[10];
    const float* Wkvp   = (const float*)d_in[11];
    const float* Wout   = (const float*)d_in[12];
    const float* b_out  = (const float*)d_in[13];
    const float* head_w = (const float*)d_in[14];

    float* out   = (float*)d_out;
    float* a_out = out + (size_t)B_ * LD_ * CS_;          // attention tensor region

    // workspace layout (floats)
    float* ws = (float*)d_ws;
    size_t off = 0;
    float* Qraw   = ws + off; off += (size_t)B_ * LD_ * 192;      // s_dst @ Wq
    float* KVraw  = ws + off; off += (size_t)B_ * LS_ * 384;      // s_src @ Wkv
    float* QPraw  = ws + off; off += (size_t)B_ * LD_ * 144;      // s_dst @ Wqp
    float* KVPraw = ws + off; off += (size_t)B_ * LS_ * 432;      // s_src @ Wkvp
    float* Qext   = ws + off; off += (size_t)B_ * H_ * LD_ * 32;
    float* KextT  = ws + off; off += (size_t)B_ * H_ * 32 * LS_;
    float* VC     = ws + off; off += (size_t)B_ * H_ * LS_ * 48;
    float* OC     = ws + off; off += (size_t)B_ * H_ * LD_ * 48;
    float* Cat    = ws + off; off += (size_t)B_ * LD_ * 576;

    const int MB = B_ * LD_;   // 2048 rows for projections
    dim3 blk32(32);

    // ---- projections (single-batch WMMA GEMMs) ----
    // 32x32 register tiles for N % 32 == 0, 16x48 tiles otherwise.
    hipLaunchKernelGGL(HIP_KERNEL_NAME(wmma_gemm_f32<2, 2>),
                       dim3(192 / 32, MB / 32, 1), blk32, 0, stream,
                       s_dst, CS_, 0LL, Wq, 192, 0LL, Qraw, 192, 0LL, CS_, nullptr);
    hipLaunchKernelGGL(HIP_KERNEL_NAME(wmma_gemm_f32<2, 2>),
                       dim3(384 / 32, MB / 32, 1), blk32, 0, stream,
                       s_src, CS_, 0LL, Wkv, 384, 0LL, KVraw, 384, 0LL, CS_, nullptr);
    hipLaunchKernelGGL(HIP_KERNEL_NAME(wmma_gemm_f32<1, 3>),
                       dim3(144 / 48, MB / 16, 1), blk32, 0, stream,
                       s_dst, CS_, 0LL, Wqp, 144, 0LL, QPraw, 144, 0LL, CS_, nullptr);
    hipLaunchKernelGGL(HIP_KERNEL_NAME(wmma_gemm_f32<1, 3>),
                       dim3(432 / 48, MB / 16, 1), blk32, 0, stream,
                       s_src, CS_, 0LL, Wkvp, 432, 0LL, KVPraw, 432, 0LL, CS_, nullptr);

    // ---- build extended Q/K and concatenated V ----
    int nQ = B_ * LD_ * H_;
    hipLaunchKernelGGL(build_q_kernel, dim3((nQ + 255) / 256), dim3(256), 0, stream,
                       Qraw, QPraw, R_dst, t_dst, head_w, Qext);
    int nK = B_ * LS_ * H_;
    hipLaunchKernelGGL(build_k_kernel, dim3((nK + 255) / 256), dim3(256), 0, stream,
                       KVraw, KVPraw, R_src, t_src, head_w, KextT, VC);

    // ---- batched logits GEMM: (512x32)x(32x512) per (b,h), straight into a_out ----
    hipLaunchKernelGGL(HIP_KERNEL_NAME(wmma_gemm_f32<2, 2>),
                       dim3(LS_ / 32, LD_ / 32, B_ * H_), blk32, 0, stream,
                       Qext, 32, (long long)LD_ * 32,
                       KextT, LS_, (long long)32 * LS_,
                       a_out, LS_, (long long)LD_ * LS_,
                       32, nullptr);

    // ---- softmax in place on a_out ----
    hipLaunchKernelGGL(softmax_kernel, dim3(B_ * H_ * LD_), dim3(256), 0, stream,
                       a_out, dst_mask, src_mask);

    // ---- batched output GEMM: a (512x512) @ [v|v_pts] (512x48) ----
    hipLaunchKernelGGL(HIP_KERNEL_NAME(wmma_gemm_f32<1, 3>),
                       dim3(1, LD_ / 16, B_ * H_), blk32, 0, stream,
                       a_out, LS_, (long long)LD_ * LS_,
                       VC, 48, (long long)LS_ * 48,
                       OC, 48, (long long)LD_ * 48,
                       LS_, nullptr);

    // ---- inverse rotation, norms, concat ----
    hipLaunchKernelGGL(build_cat_kernel, dim3((nQ + 255) / 256), dim3(256), 0, stream,
                       OC, R_dst, t_dst, Cat);

    // ---- final projection: cat (2048x576) @ Wout (576x384) + b_out ----
    hipLaunchKernelGGL(HIP_KERNEL_NAME(wmma_gemm_f32<2, 2>),
                       dim3(CS_ / 32, MB / 32, 1), blk32, 0, stream,
                       Cat, 576, 0LL, Wout, CS_, 0LL, out, CS_, 0LL, 576, b_out);
}